// Proposed_model_69131793596872
// MI455X (gfx1250) — compile-verified
//
#include <hip/hip_runtime.h>
#include <hip/hip_bf16.h>

#define NU 100000
#define NI 30000
#define DD 128

#define LANES 32
#define WPB   8           // waves per block (wave32)
#define BLOCK (LANES*WPB)
#define CHUNK 32          // edges staged per wave per step

// ---------- CDNA5 async-to-LDS path (guarded; fallback keeps it compiling) ----------
typedef __attribute__((address_space(1))) int gint_t;   // global int
typedef __attribute__((address_space(3))) int lint_t;   // LDS int

#if __has_builtin(__builtin_amdgcn_global_load_async_to_lds_b32)
#define HAVE_ASYNC_LDS 1
#else
#define HAVE_ASYNC_LDS 0
#endif

__device__ __forceinline__ void async_cp_b32(void* lds_dst, const void* g_src) {
#if HAVE_ASYNC_LDS
  __builtin_amdgcn_global_load_async_to_lds_b32(
      (gint_t*)(void*)g_src, (lint_t*)(void*)lds_dst, 0, 0);
#else
  *(int*)lds_dst = *(const int*)g_src;   // sync fallback: global load + ds store
#endif
}

#if __has_builtin(__builtin_amdgcn_s_wait_asynccnt)
#define WAIT_ASYNCCNT(n) __builtin_amdgcn_s_wait_asynccnt(n)
#else
#define WAIT_ASYNCCNT(n) asm volatile("s_wait_asynccnt %0" :: "n"(n) : "memory")
#endif

// ---------- degree / norm ----------
__global__ void degree_kernel(const int* __restrict__ ou, const int* __restrict__ oi,
                              const int* __restrict__ su, const int* __restrict__ si,
                              const int* __restrict__ du, const int* __restrict__ di,
                              float* __restrict__ degs, int nE) {
  int e = blockIdx.x * blockDim.x + threadIdx.x;
  if (e >= nE) return;
  atomicAdd(&degs[ou[e]], 1.0f);                      // oriU   [0, NU)
  atomicAdd(&degs[NU + oi[e]], 1.0f);                 // oriI
  atomicAdd(&degs[NU + NI + su[e]], 1.0f);            // seekU
  atomicAdd(&degs[2*NU + NI + si[e]], 1.0f);          // seekI
  atomicAdd(&degs[2*NU + 2*NI + du[e]], 1.0f);        // dnU
  atomicAdd(&degs[3*NU + 2*NI + di[e]], 1.0f);        // dnI
}

__global__ void norm_kernel(float* __restrict__ degs, int n) {
  int t = blockIdx.x * blockDim.x + threadIdx.x;
  if (t >= n) return;
  float d = degs[t];
  degs[t] = (d > 0.0f) ? rsqrtf(d) : 0.0f;            // deg>=1 when >0, so max(d,1)==d
}

// ---------- fused bidirectional edge pass ----------
// For edge (u,i): c = nrmU[u]*nrmI[i]
//   au[u,:] += wU * (USE_W ? w_e : 1) * c * xi[i,:]
//   ai[i,:] += wI * c * xu[u,:]
template <bool USE_W>
__global__ __launch_bounds__(BLOCK)
void edge_pass(const float* __restrict__ xu, const float* __restrict__ xi,
               float* __restrict__ au, float* __restrict__ ai,
               const int* __restrict__ eu, const int* __restrict__ ei,
               const float* __restrict__ ew,
               const float* __restrict__ nrmU, const float* __restrict__ nrmI,
               float wU, float wI, int nE) {
  __shared__ int   sU[2][WPB][CHUNK];
  __shared__ int   sI[2][WPB][CHUNK];
  __shared__ float sW[2][WPB][CHUNK];

  const int lane = threadIdx.x & (LANES - 1);
  const int wid  = threadIdx.x >> 5;
  const int gw   = blockIdx.x * WPB + wid;
  const int nW   = gridDim.x * WPB;
  const int nChunks = (nE + CHUNK - 1) / CHUNK;
  if (gw >= nChunks) return;

  auto stage = [&](int chk, int p) {
    int e = chk * CHUNK + lane;
    if (e < nE) {
      async_cp_b32(&sU[p][wid][lane], &eu[e]);
      async_cp_b32(&sI[p][wid][lane], &ei[e]);
      if constexpr (USE_W) async_cp_b32(&sW[p][wid][lane], &ew[e]);
    }
  };

  auto do_edge = [&](int p, int k) {
    int u = __builtin_amdgcn_readfirstlane(sU[p][wid][k]);
    int i = __builtin_amdgcn_readfirstlane(sI[p][wid][k]);
    float we = 1.0f;
    if constexpr (USE_W) we = sW[p][wid][k];
    const float cc = nrmU[u] * nrmI[i];
    const float4 vi = *(const float4*)(xi + (size_t)i * DD + 4 * lane);
    const float4 vu = *(const float4*)(xu + (size_t)u * DD + 4 * lane);
    const float fU = wU * cc * we;
    const float fI = wI * cc;
    float* pu = au + (size_t)u * DD + 4 * lane;
    float* pi = ai + (size_t)i * DD + 4 * lane;
    atomicAdd(pu + 0, vi.x * fU); atomicAdd(pu + 1, vi.y * fU);
    atomicAdd(pu + 2, vi.z * fU); atomicAdd(pu + 3, vi.w * fU);
    atomicAdd(pi + 0, vu.x * fI); atomicAdd(pi + 1, vu.y * fI);
    atomicAdd(pi + 2, vu.z * fI); atomicAdd(pi + 3, vu.w * fI);
  };

  int p = 0;
  stage(gw, p);                                   // prologue: stage first chunk
  for (int chk = gw; chk < nChunks; chk += nW) {
    const int nxt = chk + nW;
    if (nxt < nChunks) {
      stage(nxt, p ^ 1);                          // overlap next chunk's DMA
      WAIT_ASYNCCNT(USE_W ? 3 : 2);               // current chunk done; next may be in flight
    } else {
      WAIT_ASYNCCNT(0);
    }

    const int baseE = chk * CHUNK;
    if (nE - baseE >= CHUNK) {
      // hot path: fixed trip count, unroll x2 => two independent row-load pairs in flight
      #pragma unroll 2
      for (int k = 0; k < CHUNK; ++k) do_edge(p, k);
    } else {
      const int cnt = nE - baseE;
      for (int k = 0; k < cnt; ++k) do_edge(p, k);
    }
    p ^= 1;
  }
}

extern "C" void kernel_launch(void* const* d_in, const int* in_sizes, int n_in,
                              void* d_out, int out_size, void* d_ws, size_t ws_size,
                              hipStream_t stream) {
  const float* user_emb = (const float*)d_in[0];
  const float* item_emb = (const float*)d_in[1];
  const float* wedge    = (const float*)d_in[2];
  const int* ou = (const int*)d_in[3];
  const int* oi = (const int*)d_in[4];
  const int* su = (const int*)d_in[5];
  const int* si = (const int*)d_in[6];
  const int* du = (const int*)d_in[7];
  const int* di = (const int*)d_in[8];
  const int E = in_sizes[2];

  const size_t SU = (size_t)NU * DD, SI = (size_t)NI * DD;
  float* out = (float*)d_out;
  float* hu0 = out;                      // slot0: hu ping      (final hu)
  float* hi0 = out + SU;                 // slot1: hi ping      (final hi)
  float* hu1 = out + SU + SI;            // slot2: hu pong / seekf_u / final dup
  float* hi1 = out + 2 * SU + SI;        // slot3: hi pong / seekf_i / final dup
  float* h1u = out + 2 * SU + 2 * SI;    // slot4: h1u
  float* h1i = out + 3 * SU + 2 * SI;    // slot5: h1i

  float* degs = (float*)d_ws;            // 3*(NU+NI) floats: degrees -> norms (in place)
  const int NNORM = 3 * (NU + NI);
  float* t1u = degs + NNORM;             // branch-2 ping-pong temp
  float* t1i = t1u + SU;

  const float* nOriU  = degs;
  const float* nOriI  = degs + NU;
  const float* nSeekU = degs + NU + NI;
  const float* nSeekI = degs + 2 * NU + NI;
  const float* nDnU   = degs + 2 * NU + 2 * NI;
  const float* nDnI   = degs + 3 * NU + 2 * NI;

  // degrees -> norms
  (void)hipMemsetAsync(degs, 0, sizeof(float) * NNORM, stream);
  degree_kernel<<<(E + 255) / 256, 256, 0, stream>>>(ou, oi, su, si, du, di, degs, E);
  norm_kernel<<<(NNORM + 255) / 256, 256, 0, stream>>>(degs, NNORM);

  const int nChunks = (E + CHUNK - 1) / CHUNK;
  int grid = (nChunks + WPB - 1) / WPB;
  if (grid > 4096) grid = 4096;

  auto zero2 = [&](float* a, float* b) {
    (void)hipMemsetAsync(a, 0, SU * sizeof(float), stream);
    (void)hipMemsetAsync(b, 0, SI * sizeof(float), stream);
  };
  auto copy2 = [&](float* dU, float* dI, const float* sU_, const float* sI_) {
    (void)hipMemcpyAsync(dU, sU_, SU * sizeof(float), hipMemcpyDeviceToDevice, stream);
    (void)hipMemcpyAsync(dI, sI_, SI * sizeof(float), hipMemcpyDeviceToDevice, stream);
  };

  // ---------------- branch 1: ori + seek, 3 layers ----------------
  // L1: inputs -> slot0/1
  zero2(hu0, hi0);
  edge_pass<false><<<grid, BLOCK, 0, stream>>>(user_emb, item_emb, hu0, hi0, ou, oi, nullptr, nOriU, nOriI, 1.0f, 1.0f, E);
  edge_pass<true ><<<grid, BLOCK, 0, stream>>>(user_emb, item_emb, hu0, hi0, su, si, wedge,  nSeekU, nSeekI, 0.5f, 0.5f, E);
  // L2: slot0/1 -> slot2/3
  zero2(hu1, hi1);
  edge_pass<false><<<grid, BLOCK, 0, stream>>>(hu0, hi0, hu1, hi1, ou, oi, nullptr, nOriU, nOriI, 1.0f, 1.0f, E);
  edge_pass<true ><<<grid, BLOCK, 0, stream>>>(hu0, hi0, hu1, hi1, su, si, wedge,  nSeekU, nSeekI, 0.5f, 0.5f, E);
  // L3: slot2/3 -> slot0/1  (final hu,hi)
  zero2(hu0, hi0);
  edge_pass<false><<<grid, BLOCK, 0, stream>>>(hu1, hi1, hu0, hi0, ou, oi, nullptr, nOriU, nOriI, 1.0f, 1.0f, E);
  edge_pass<true ><<<grid, BLOCK, 0, stream>>>(hu1, hi1, hu0, hi0, su, si, wedge,  nSeekU, nSeekI, 0.5f, 0.5f, E);

  // seek-final terms (x0.5 folded) from final h into slot2/3
  zero2(hu1, hi1);
  edge_pass<true ><<<grid, BLOCK, 0, stream>>>(hu0, hi0, hu1, hi1, su, si, wedge, nSeekU, nSeekI, 0.5f, 0.5f, E);

  // ---------------- branch 2: dn + constant seekf, 3 layers ----------------
  // L1: inputs -> slot4/5 (seeded with seekf)
  copy2(h1u, h1i, hu1, hi1);
  edge_pass<false><<<grid, BLOCK, 0, stream>>>(user_emb, item_emb, h1u, h1i, du, di, nullptr, nDnU, nDnI, 1.0f, 1.0f, E);
  // L2: slot4/5 -> ws temp
  copy2(t1u, t1i, hu1, hi1);
  edge_pass<false><<<grid, BLOCK, 0, stream>>>(h1u, h1i, t1u, t1i, du, di, nullptr, nDnU, nDnI, 1.0f, 1.0f, E);
  // L3: ws temp -> slot4/5 (final h1u,h1i)
  copy2(h1u, h1i, hu1, hi1);
  edge_pass<false><<<grid, BLOCK, 0, stream>>>(t1u, t1i, h1u, h1i, du, di, nullptr, nDnU, nDnI, 1.0f, 1.0f, E);

  // outputs 3,4 duplicate (h_sub1 aliases h): slot0/1 -> slot2/3
  copy2(hu1, hi1, hu0, hi0);
}